// SpikeFP32GELU_29875792511548
// MI455X (gfx1250) — compile-verified
//
#include <hip/hip_runtime.h>
#include <hip/hip_bf16.h>
#include <stdint.h>
#include <math.h>

// SpikeFP32GELU for MI455X (gfx1250), compile-only loop.
//
// Bit-pulse fp32 decode -> fp64 GELU -> single-round fp32 -> bit-pulse encode.
// Pure streaming kernel (512 MiB @ 23.3 TB/s ~= 23 us floor). Input staging via
// the CDNA5 Tensor Data Mover (tensor_load_to_lds + TENSORcnt) with hardware
// pad-insertion producing a bank-conflict-minimal LDS layout; output drain via
// async LDS->global stores (ASYNCcnt) with non-temporal hint. wave32-native.

#define TILE    256   // values per workgroup (full-tile kernel: no guards)
#define PAD_DW  36    // dwords per value slot in LDS (32 data + 4 pad)

typedef unsigned int u32x4 __attribute__((ext_vector_type(4)));
typedef int          i32x8 __attribute__((ext_vector_type(8)));

__device__ __forceinline__ unsigned gelu_word(unsigned w) {
    // GELU(x) ~= x * sigmoid(1.702 x), fp64 intermediates, one fp32 rounding.
    float  xf = __uint_as_float(w);
    double xd = (double)xf;
    double yd = xd / (1.0 + exp(-1.702 * xd));
    float  yf = (float)yd;
    // Reference encodes both +0 and -0 as the all-zero pulse vector.
    return (yf == 0.0f) ? 0u : __float_as_uint(yf);
}

__global__ __launch_bounds__(256) void spike_gelu_kernel(const float* __restrict__ x,
                                                         float* __restrict__ out) {
    __shared__ __align__(16) float smem[TILE * PAD_DW];   // 36,864 bytes

    const int tid = threadIdx.x;
    const long long tile_base = (long long)blockIdx.x * TILE;

    // LDS byte offset of smem within this workgroup's allocation.
    const unsigned lds_base = (unsigned)(uintptr_t)(&smem[0]);

    // ---- Stage 1: TDM load of the whole 32 KB tile, global -> LDS ----------
    // 2D tensor: 256 rows (values) x 32 floats. pad_enable inserts 4 dwords of
    // LDS padding after every 32 dwords stored -> 36-dword slot stride, which
    // makes the per-thread B128 reads below 2-way-conflict-minimal on 64 banks.
    if (tid < 32) {                       // wave 0 only (wave-uniform branch)
        unsigned long long ga =
            (unsigned long long)(uintptr_t)(x + tile_base * 32);
        u32x4 g0;
        g0.x = 1u;                                        // count=1 (valid user D#)
        g0.y = lds_base;                                  // lds_addr (bytes)
        g0.z = (unsigned)(ga & 0xFFFFFFFFu);              // global_addr[31:0]
        g0.w = (unsigned)((ga >> 32) & 0x01FFFFFFu)       // global_addr[56:32]
             | (2u << 30);                                // type=2 ("image")
        i32x8 g1;
        g1[0] = (int)((2u << 16)    // data_size=2 -> 4-byte elements
                    | (1u << 20)    // pad_enable
                    | (4u << 22)    // pad_interval code 4 -> every 32 dwords
                    | (3u << 25));  // pad_amount  code 3 -> 4 dwords
        g1[1] = (int)(32u << 16);          // tensor_dim0 = 32 (low 16 bits)
        g1[2] = (int)((unsigned)TILE << 16); // tensor_dim0 hi=0 | tensor_dim1 lo=256
        g1[3] = (int)(32u << 16);          // tensor_dim1 hi=0 | tile_dim0 = 32
        g1[4] = (int)TILE;                 // tile_dim1 = 256 | tile_dim2 = 0
        g1[5] = 32;                        // tensor_dim0_stride = 32 (lo)
        g1[6] = 0;                         // stride0 hi | stride1 lo (unused, 2D)
        g1[7] = 0;                         // stride1 hi (unused)
        asm volatile("tensor_load_to_lds %0, %1" :: "s"(g0), "s"(g1) : "memory");
        __builtin_amdgcn_s_wait_tensorcnt(0);
    }
    __syncthreads();

    // ---- Stage 2: per-thread decode -> fp64 GELU -> encode (own slot only) -
    {
        // Decode: 32 pulses (MSB first) -> u32 word. Pulses are exactly 0.0/1.0,
        // so v_cvt_u32_f32 + v_lshl_or assemble the word (2 VALU per bit).
        unsigned w = 0u;
        #pragma unroll
        for (int r = 0; r < 8; ++r) {
            float4 q = *reinterpret_cast<const float4*>(&smem[tid * PAD_DW + r * 4]);
            w = (w << 1) | (unsigned)q.x;
            w = (w << 1) | (unsigned)q.y;
            w = (w << 1) | (unsigned)q.z;
            w = (w << 1) | (unsigned)q.w;
        }

        unsigned yw = gelu_word(w);

        // Encode: u32 word -> 32 pulses, written back into our own LDS slot.
        #pragma unroll
        for (int r = 0; r < 8; ++r) {
            float4 q;
            q.x = (float)((yw >> (31 - (4 * r + 0))) & 1u);
            q.y = (float)((yw >> (31 - (4 * r + 1))) & 1u);
            q.z = (float)((yw >> (31 - (4 * r + 2))) & 1u);
            q.w = (float)((yw >> (31 - (4 * r + 3))) & 1u);
            *reinterpret_cast<float4*>(&smem[tid * PAD_DW + r * 4]) = q;
        }
    }
    __syncthreads();

    // ---- Stage 3: async store LDS -> global, coalesced 16 B/lane, NT -------
    {
        float* dst = out + tile_base * 32;
        #pragma unroll
        for (int r = 0; r < 8; ++r) {
            int c = r * 256 + tid;           // 16-byte chunk id within tile
            unsigned lds_off = lds_base +
                (unsigned)((c >> 3) * (PAD_DW * 4) + (c & 7) * 16);
            unsigned long long gaddr =
                (unsigned long long)(uintptr_t)(dst + (size_t)c * 4);
            asm volatile("global_store_async_from_lds_b128 %0, %1, off th:TH_STORE_NT"
                         :: "v"(gaddr), "v"(lds_off) : "memory");
        }
    }
    asm volatile("s_wait_asynccnt 0" ::: "memory");
}

// Guarded scalar tail for a partial final tile (not launched for the given
// shape, where nvals % TILE == 0).
__global__ __launch_bounds__(256) void spike_gelu_tail(const float* __restrict__ x,
                                                       float* __restrict__ out,
                                                       long long start,
                                                       long long nvals) {
    long long v = start + (long long)blockIdx.x * blockDim.x + threadIdx.x;
    if (v >= nvals) return;
    const float* src = x + v * 32;
    unsigned w = 0u;
    #pragma unroll
    for (int j = 0; j < 32; ++j) w = (w << 1) | (unsigned)src[j];
    unsigned yw = gelu_word(w);
    float* dst = out + v * 32;
    #pragma unroll
    for (int j = 0; j < 32; ++j) dst[j] = (float)((yw >> (31 - j)) & 1u);
}

extern "C" void kernel_launch(void* const* d_in, const int* in_sizes, int n_in,
                              void* d_out, int out_size, void* d_ws, size_t ws_size,
                              hipStream_t stream) {
    (void)n_in; (void)out_size; (void)d_ws; (void)ws_size;
    const float* x = (const float*)d_in[0];
    float* out = (float*)d_out;
    long long nfloats = (long long)in_sizes[0];
    long long nvals = nfloats / 32;                 // 2,097,152 for the given shape
    long long full = nvals / TILE;                  // 8192 full tiles
    long long rem  = nvals - full * TILE;           // 0 for the given shape
    if (full > 0) {
        hipLaunchKernelGGL(spike_gelu_kernel, dim3((uint32_t)full), dim3(256), 0,
                           stream, x, out);
    }
    if (rem > 0) {
        hipLaunchKernelGGL(spike_gelu_tail, dim3(1), dim3(256), 0, stream,
                           x, out, full * TILE, nvals);
    }
}